// ROIGridPooler_50053548868056
// MI455X (gfx1250) — compile-verified
//
#include <hip/hip_runtime.h>
#include <math.h>

// ---- problem constants (from reference) ----
#define B_    4
#define N_    256
#define BINS_ 12
#define S_    1024
#define C_    256
#define KP_   64          // key points per ROI (G^3)
#define DIN_  259         // 3 + C
#define K1_   260         // DIN padded to multiple of 4 for f32 WMMA
#define CH_   128         // hidden channels
#define LDX_  262         // LDS row stride of X tile (even -> 8B aligned pairs)
#define LDH_  134         // LDS row stride of hidden tile

typedef __attribute__((ext_vector_type(2))) float v2f;
typedef __attribute__((ext_vector_type(8))) float v8f;

static __device__ __forceinline__ v8f wmma_f32_16x16x4(v2f a, v2f b, v8f c) {
#if defined(__gfx1250__)
  // 8 args: (neg_a, A, neg_b, B, c_mod, C, reuse_a, reuse_b)
  return __builtin_amdgcn_wmma_f32_16x16x4_f32(false, a, false, b, (short)0, c,
                                               false, false);
#else
  (void)a; (void)b;
  return c;  // host-pass placeholder, never executed
#endif
}

// ---- LDS layout for the fused kernel (units: floats) ----
constexpr int OFF_X    = 0;
constexpr int OFF_H1   = OFF_X + KP_ * LDX_;      // 16768
constexpr int OFF_SEED = OFF_H1 + KP_ * LDH_;     // 25344
constexpr int OFF_CD   = OFF_SEED + S_ * 3;       // 28416
constexpr int OFF_CI   = OFF_CD + KP_ * 4 * 3;    // 29184
constexpr int OFF_TW   = OFF_CI + KP_ * 4 * 3;    // 29952
constexpr int OFF_TI   = OFF_TW + KP_ * 3;        // 30144
constexpr int SMEM_FLOATS = OFF_TI + KP_ * 3;     // 30336 (~121.3 KB)

// =====================================================================
// Kernel 1: pad w1 to K=260 and fold BN into per-channel scale/bias
// =====================================================================
__global__ void roigrid_prep(const float* __restrict__ w1,
                             const float* __restrict__ g1, const float* __restrict__ be1,
                             const float* __restrict__ mu1, const float* __restrict__ va1,
                             const float* __restrict__ g2, const float* __restrict__ be2,
                             const float* __restrict__ mu2, const float* __restrict__ va2,
                             float* __restrict__ w1p,
                             float* __restrict__ a1, float* __restrict__ b1,
                             float* __restrict__ a2, float* __restrict__ b2) {
  int idx = blockIdx.x * 256 + threadIdx.x;
  if (idx < CH_ * K1_) {
    int r = idx / K1_, k = idx % K1_;
    w1p[idx] = (k < DIN_) ? w1[r * DIN_ + k] : 0.0f;
  } else if (idx < CH_ * K1_ + CH_) {
    int c = idx - CH_ * K1_;
    float a = g1[c] * rsqrtf(va1[c] + 1e-5f);
    a1[c] = a;
    b1[c] = be1[c] - mu1[c] * a;
  } else if (idx < CH_ * K1_ + 2 * CH_) {
    int c = idx - CH_ * K1_ - CH_;
    float a = g2[c] * rsqrtf(va2[c] + 1e-5f);
    a2[c] = a;
    b2[c] = be2[c] - mu2[c] * a;
  }
}

// =====================================================================
// Kernel 2: heading decode + 64 rotated grid key points per (b,n)
// =====================================================================
__global__ __launch_bounds__(256) void roigrid_keypoints(
    const float* __restrict__ hs, const float* __restrict__ hr,
    const float* __restrict__ rois, const float* __restrict__ vote,
    float* __restrict__ kp) {
  int t = blockIdx.x * 256 + threadIdx.x;
  if (t >= B_ * N_) return;

  const float* sc = hs + t * BINS_;
  int cls = 0;
  float best = sc[0];
  for (int i = 1; i < BINS_; ++i)
    if (sc[i] > best) { best = sc[i]; cls = i; }   // strict > == first-max, like argmax
  float res = hr[t * BINS_ + cls];

  const float TWO_PI = 6.28318530717958647692f;
  float ang = (float)cls * (TWO_PI / (float)BINS_) + res;
  float h = fmodf(ang, TWO_PI);
  if (h < 0.0f) h += TWO_PI;                       // jnp.mod semantics (sign of divisor)

  float c = cosf(h), s = sinf(h);
  const float* r = rois + t * 6;
  float lsx = r[0] + r[3], lsy = r[1] + r[4], lsz = r[2] + r[5];
  float clx = r[0], cly = r[1], clz = r[2];
  float cx = vote[t * 3 + 0], cy = vote[t * 3 + 1], cz = vote[t * 3 + 2];

  for (int i = 0; i < 4; ++i)
    for (int j = 0; j < 4; ++j)
      for (int k = 0; k < 4; ++k) {
        float px = (i + 0.5f) * 0.25f * lsx - clx;
        float py = (j + 0.5f) * 0.25f * lsy - cly;
        float pz = (k + 0.5f) * 0.25f * lsz - clz;
        // einsum("bkc,bcd->bkd", pts, rotz): out_d = sum_c pts_c * R[c][d]
        float rx =  px * c + py * s;
        float ry = -px * s + py * c;
        int m = t * KP_ + (i * 16 + j * 4 + k);
        kp[m * 3 + 0] = rx + cx;
        kp[m * 3 + 1] = ry + cy;
        kp[m * 3 + 2] = pz + cz;
      }
}

// =====================================================================
// Kernel 3: fused top3 + interp + WMMA MLP (259->128->128) + maxpool
// One workgroup per (b,n): 64-row M tile, 256 threads = 8 wave32s.
// =====================================================================
__global__ __launch_bounds__(256) void roigrid_main(
    const float* __restrict__ vote, const float* __restrict__ seed_xyz,
    const float* __restrict__ seed_feat, const float* __restrict__ w1p,
    const float* __restrict__ a1, const float* __restrict__ b1,
    const float* __restrict__ w2, const float* __restrict__ a2,
    const float* __restrict__ b2, const float* __restrict__ kp,
    float* __restrict__ out) {
  extern __shared__ float smem[];
  float* Xs    = smem + OFF_X;     // 64 x 260 (stride 262): [rel(3) | interp(256) | pad]
  float* H1    = smem + OFF_H1;    // 64 x 128 (stride 134)
  float* seeds = smem + OFF_SEED;  // 1024 x 3
  float* cd    = smem + OFF_CD;    // partial top3 dists  [64][4][3]
  int*   ci    = (int*)(smem + OFF_CI);
  float* tw    = smem + OFF_TW;    // final weights [64][3]
  int*   ti    = (int*)(smem + OFF_TI);

  const int bn  = blockIdx.x;
  const int b   = bn / N_;
  const int n   = bn % N_;
  const int m0  = bn * KP_;
  const int tid = threadIdx.x;

  // ---- phase 0: stage seed_xyz into LDS ----
  for (int i = tid; i < S_ * 3; i += 256) seeds[i] = seed_xyz[b * S_ * 3 + i];
  __syncthreads();

  // ---- phase 1: partial top-3 (4 threads per point, 256 seeds each) ----
  {
    int p = tid >> 2, q = tid & 3;
    float px = kp[(m0 + p) * 3 + 0];
    float py = kp[(m0 + p) * 3 + 1];
    float pz = kp[(m0 + p) * 3 + 2];
    float d0 = 3.4e38f, d1v = 3.4e38f, d2v = 3.4e38f;
    int i0 = 0, i1 = 0, i2 = 0;
    int sbeg = q * 256;
    for (int s = sbeg; s < sbeg + 256; ++s) {
      float dx = px - seeds[s * 3 + 0];
      float dy = py - seeds[s * 3 + 1];
      float dz = pz - seeds[s * 3 + 2];
      float d = dx * dx + dy * dy + dz * dz;
      if (d < d2v) {
        if (d < d0)        { d2v = d1v; i2 = i1; d1v = d0; i1 = i0; d0 = d; i0 = s; }
        else if (d < d1v)  { d2v = d1v; i2 = i1; d1v = d;  i1 = s; }
        else               { d2v = d;  i2 = s; }
      }
    }
    int base = (p * 4 + q) * 3;
    cd[base + 0] = d0;  cd[base + 1] = d1v; cd[base + 2] = d2v;
    ci[base + 0] = i0;  ci[base + 1] = i1;  ci[base + 2] = i2;
  }
  __syncthreads();

  // ---- phase 1b: merge partials, weights, rel coords (waves 0-1 fully active) ----
  if (tid < KP_) {
    int p = tid;
    float d0 = 3.4e38f, d1v = 3.4e38f, d2v = 3.4e38f;
    int i0 = 0, i1 = 0, i2 = 0;
    for (int q = 0; q < 4; ++q)
      for (int j = 0; j < 3; ++j) {
        float d = cd[(p * 4 + q) * 3 + j];
        int   s = ci[(p * 4 + q) * 3 + j];
        if (d < d2v) {
          if (d < d0)       { d2v = d1v; i2 = i1; d1v = d0; i1 = i0; d0 = d; i0 = s; }
          else if (d < d1v) { d2v = d1v; i2 = i1; d1v = d;  i1 = s; }
          else              { d2v = d;  i2 = s; }
        }
      }
    float t0 = sqrtf(fmaxf(d0, 0.0f));
    float t1 = sqrtf(fmaxf(d1v, 0.0f));
    float t2 = sqrtf(fmaxf(d2v, 0.0f));
    float w0 = 1.0f / (t0 + 1e-8f);
    float w1v = 1.0f / (t1 + 1e-8f);
    float w2v = 1.0f / (t2 + 1e-8f);
    float wsum = w0 + w1v + w2v;
    tw[p * 3 + 0] = w0 / wsum;
    tw[p * 3 + 1] = w1v / wsum;
    tw[p * 3 + 2] = w2v / wsum;
    ti[p * 3 + 0] = i0; ti[p * 3 + 1] = i1; ti[p * 3 + 2] = i2;
    // rel = key_point - vote center ; zero the K padding columns
    float cx = vote[(b * N_ + n) * 3 + 0];
    float cy = vote[(b * N_ + n) * 3 + 1];
    float cz = vote[(b * N_ + n) * 3 + 2];
    Xs[p * LDX_ + 0] = kp[(m0 + p) * 3 + 0] - cx;
    Xs[p * LDX_ + 1] = kp[(m0 + p) * 3 + 1] - cy;
    Xs[p * LDX_ + 2] = kp[(m0 + p) * 3 + 2] - cz;
    Xs[p * LDX_ + 259] = 0.0f;
    Xs[p * LDX_ + 260] = 0.0f;
    Xs[p * LDX_ + 261] = 0.0f;
  }
  __syncthreads();

  // ---- phase 2: 3-NN feature interpolation into Xs[:,3:259] ----
  for (int e = tid; e < KP_ * C_; e += 256) {
    int p = e >> 8;           // C_ == 256
    int c = e & 255;
    const float* F = seed_feat + ((size_t)b * C_ + c) * S_;
    float v = tw[p * 3 + 0] * F[ti[p * 3 + 0]] +
              tw[p * 3 + 1] * F[ti[p * 3 + 1]] +
              tw[p * 3 + 2] * F[ti[p * 3 + 2]];
    Xs[p * LDX_ + 3 + c] = v;
  }
  __syncthreads();

  // ---- wave tiling for WMMA: wave w -> M strip (w>>1)*16, N half (w&1)*64 ----
  const int lane   = tid & 31;
  const int wv     = tid >> 5;
  const int mstrip = wv >> 1;
  const int nhalf  = wv & 1;
  const int am     = mstrip * 16 + (lane & 15);  // A row for this lane
  const int koff   = (lane >> 4) * 2;            // K sub-offset {0,2}
  const int nlane  = lane & 15;
  const int nc0 = (nhalf * 4 + 0) * 16 + nlane;
  const int nc1 = (nhalf * 4 + 1) * 16 + nlane;
  const int nc2 = (nhalf * 4 + 2) * 16 + nlane;
  const int nc3 = (nhalf * 4 + 3) * 16 + nlane;
  const int mbase = mstrip * 16 + (lane >> 4) * 8;  // D rows: v + 8*(lane>=16)

  // ---- phase 3: GEMM1  H1 = relu(a1 * (X @ w1p^T) + b1) ----
  {
    v8f acc0 = {}, acc1 = {}, acc2 = {}, acc3 = {};
    for (int k = 0; k < K1_; k += 4) {
      v2f a  = *(const v2f*)&Xs[am * LDX_ + k + koff];
      v2f f0 = *(const v2f*)&w1p[nc0 * K1_ + k + koff];
      v2f f1 = *(const v2f*)&w1p[nc1 * K1_ + k + koff];
      v2f f2 = *(const v2f*)&w1p[nc2 * K1_ + k + koff];
      v2f f3 = *(const v2f*)&w1p[nc3 * K1_ + k + koff];
      acc0 = wmma_f32_16x16x4(a, f0, acc0);
      acc1 = wmma_f32_16x16x4(a, f1, acc1);
      acc2 = wmma_f32_16x16x4(a, f2, acc2);
      acc3 = wmma_f32_16x16x4(a, f3, acc3);
    }
    float s0 = a1[nc0], o0 = b1[nc0];
    float s1 = a1[nc1], o1 = b1[nc1];
    float s2 = a1[nc2], o2 = b1[nc2];
    float s3 = a1[nc3], o3 = b1[nc3];
    for (int v = 0; v < 8; ++v) {
      int m = mbase + v;
      H1[m * LDH_ + nc0] = fmaxf(acc0[v] * s0 + o0, 0.0f);
      H1[m * LDH_ + nc1] = fmaxf(acc1[v] * s1 + o1, 0.0f);
      H1[m * LDH_ + nc2] = fmaxf(acc2[v] * s2 + o2, 0.0f);
      H1[m * LDH_ + nc3] = fmaxf(acc3[v] * s3 + o3, 0.0f);
    }
  }
  __syncthreads();

  // ---- phase 4: GEMM2  H2 = relu(a2 * (H1 @ w2^T) + b2), H2 reuses Xs ----
  {
    v8f acc0 = {}, acc1 = {}, acc2 = {}, acc3 = {};
    for (int k = 0; k < CH_; k += 4) {
      v2f a  = *(const v2f*)&H1[am * LDH_ + k + koff];
      v2f f0 = *(const v2f*)&w2[nc0 * CH_ + k + koff];
      v2f f1 = *(const v2f*)&w2[nc1 * CH_ + k + koff];
      v2f f2 = *(const v2f*)&w2[nc2 * CH_ + k + koff];
      v2f f3 = *(const v2f*)&w2[nc3 * CH_ + k + koff];
      acc0 = wmma_f32_16x16x4(a, f0, acc0);
      acc1 = wmma_f32_16x16x4(a, f1, acc1);
      acc2 = wmma_f32_16x16x4(a, f2, acc2);
      acc3 = wmma_f32_16x16x4(a, f3, acc3);
    }
    float s0 = a2[nc0], o0 = b2[nc0];
    float s1 = a2[nc1], o1 = b2[nc1];
    float s2 = a2[nc2], o2 = b2[nc2];
    float s3 = a2[nc3], o3 = b2[nc3];
    for (int v = 0; v < 8; ++v) {
      int m = mbase + v;
      Xs[m * LDH_ + nc0] = fmaxf(acc0[v] * s0 + o0, 0.0f);
      Xs[m * LDH_ + nc1] = fmaxf(acc1[v] * s1 + o1, 0.0f);
      Xs[m * LDH_ + nc2] = fmaxf(acc2[v] * s2 + o2, 0.0f);
      Xs[m * LDH_ + nc3] = fmaxf(acc3[v] * s3 + o3, 0.0f);
    }
  }
  __syncthreads();

  // ---- phase 5: maxpool over the 64 key points, write out[b, c, n] ----
  if (tid < CH_) {  // waves 0-3 fully active
    float mx = -3.4e38f;
    for (int p = 0; p < KP_; ++p) mx = fmaxf(mx, Xs[p * LDH_ + tid]);
    out[((size_t)b * CH_ + tid) * N_ + n] = mx;
  }
}

// =====================================================================
extern "C" void kernel_launch(void* const* d_in, const int* in_sizes, int n_in,
                              void* d_out, int out_size, void* d_ws, size_t ws_size,
                              hipStream_t stream) {
  (void)in_sizes; (void)n_in; (void)out_size; (void)ws_size;

  const float* hs       = (const float*)d_in[0];
  const float* hr       = (const float*)d_in[1];
  const float* rois     = (const float*)d_in[2];
  const float* vote     = (const float*)d_in[3];
  const float* seed_xyz = (const float*)d_in[4];
  const float* seed_ft  = (const float*)d_in[5];
  const float* w1       = (const float*)d_in[6];
  const float* g1       = (const float*)d_in[7];
  const float* be1      = (const float*)d_in[8];
  const float* mu1      = (const float*)d_in[9];
  const float* va1      = (const float*)d_in[10];
  const float* w2       = (const float*)d_in[11];
  const float* g2       = (const float*)d_in[12];
  const float* be2      = (const float*)d_in[13];
  const float* mu2      = (const float*)d_in[14];
  const float* va2      = (const float*)d_in[15];
  float* out = (float*)d_out;

  // workspace layout (floats): kp | w1p | a1 | b1 | a2 | b2  (~0.92 MB)
  float* ws  = (float*)d_ws;
  float* kp  = ws;
  float* w1p = kp + (size_t)B_ * N_ * KP_ * 3;
  float* a1  = w1p + CH_ * K1_;
  float* b1  = a1 + CH_;
  float* a2  = b1 + CH_;
  float* b2  = a2 + CH_;

  int prep_elems = CH_ * K1_ + 2 * CH_;
  roigrid_prep<<<(prep_elems + 255) / 256, 256, 0, stream>>>(
      w1, g1, be1, mu1, va1, g2, be2, mu2, va2, w1p, a1, b1, a2, b2);

  roigrid_keypoints<<<(B_ * N_ + 255) / 256, 256, 0, stream>>>(hs, hr, rois, vote, kp);

  size_t smem = (size_t)SMEM_FLOATS * sizeof(float);
  (void)hipFuncSetAttribute((const void*)roigrid_main,
                            hipFuncAttributeMaxDynamicSharedMemorySize, (int)smem);
  roigrid_main<<<B_ * N_, 256, smem, stream>>>(
      vote, seed_xyz, seed_ft, w1p, a1, b1, w2, a2, b2, kp, out);
}